// Loss_84215718740601
// MI455X (gfx1250) — compile-verified
//
#include <hip/hip_runtime.h>
#include <hip/hip_bf16.h>
#include <math.h>

// ---------------------------------------------------------------------------
// Loss kernel for MI455X (gfx1250, wave32).
// Strategy: only intra-class pairwise distances matter (~64 samples/class),
// so bucket by class and compute per-class 16x16 Gram tiles with
// v_wmma_f32_16x16x4_f32 (full fp32 precision, matching the fp32 reference).
// Total useful work ~0.13 GFLOP + one 4MB pass over y: latency-bound,
// L2-resident on the 192MB L2.
// ---------------------------------------------------------------------------

#define NCLS 32
#define MAXM 2048          // worst-case members per class (bucket capacity)
#define KTOP 2

typedef __attribute__((ext_vector_type(2))) float v2f;
typedef __attribute__((ext_vector_type(8))) float v8f;

#define NEGBIG (-1.0e30f)

// --- kernel 1: bucket samples by class (1 wave, deterministic order) -------
__global__ void bucket_kernel(const int* __restrict__ tgt,
                              int* __restrict__ counts,
                              int* __restrict__ members,
                              int B) {
    int c = threadIdx.x;
    if (c >= NCLS) return;
    int base = c * MAXM;
    int m = 0;
    for (int i = 0; i < B; ++i) {
        if (tgt[i] == c) {
            if (m < MAXM) members[base + m] = i;
            ++m;
        }
    }
    if (m > MAXM) m = MAXM;
    counts[c] = m;
    // pad to a whole number of 16-row tiles with a safe (valid) row index
    int fill = (m > 0) ? members[base] : 0;
    int padTo = (m + 15) & ~15;
    for (int s = m; s < padTo; ++s) members[base + s] = fill;
}

// --- kernel 2: per-row squared norms (wave per row) ------------------------
__global__ __launch_bounds__(256) void rownorm_kernel(const float* __restrict__ y,
                                                      float* __restrict__ sq,
                                                      int B, int D) {
    int w    = (int)((blockIdx.x * blockDim.x + threadIdx.x) >> 5);
    int lane = threadIdx.x & 31;
    if (w >= B) return;
    const float* row = y + (size_t)w * D;
    float s = 0.0f;
    for (int d = lane; d < D; d += 32) {
        float v = row[d];
        s = fmaf(v, v, s);
    }
    for (int off = 16; off > 0; off >>= 1) s += __shfl_xor(s, off, 32);
    if (lane == 0) sq[w] = s;
}

// --- kernel 3: per-class centers (block per class) -------------------------
__global__ __launch_bounds__(256) void centers_kernel(const float* __restrict__ y,
                                                      const int* __restrict__ members,
                                                      const int* __restrict__ counts,
                                                      float* __restrict__ centers,
                                                      int D) {
    int c = blockIdx.x;
    int m = counts[c];
    float inv = 1.0f / fmaxf((float)m, 1.0f);
    int base = c * MAXM;
    for (int col = threadIdx.x; col < D; col += blockDim.x) {
        float acc = 0.0f;
        for (int s = 0; s < m; ++s) {
            int g = members[base + s];
            acc += y[(size_t)g * D + col];
        }
        centers[(size_t)c * D + col] = acc * inv;
    }
}

// --- kernel 4: per-class top-2 intra-class distances via f32 WMMA ----------
// Block per class, wave per 16x16 tile pair; K stepped by 4 with
// v_wmma_f32_16x16x4_f32. A: lane<16 -> (row=lane, K={k0,k0+1}),
// lane>=16 -> (row=lane-16, K={k0+2,k0+3}); B mirrors it, so both operands
// are the same float2 pattern from their tile's rows (Gram structure).
__global__ __launch_bounds__(256) void intra_topk_kernel(const float* __restrict__ y,
                                                         const int* __restrict__ members,
                                                         const int* __restrict__ counts,
                                                         const float* __restrict__ sq,
                                                         float* __restrict__ terms,
                                                         int D) {
    const int c      = blockIdx.x;
    const int m      = counts[c];
    const int cbase  = c * MAXM;
    const int ntiles = (m + 15) >> 4;
    const int npairs = ntiles * (ntiles + 1) / 2;

    const int tid    = threadIdx.x;
    const int wid    = tid >> 5;
    const int lane   = tid & 31;
    const int half   = lane >> 4;   // 0: K+{0,1}, 1: K+{2,3}
    const int lm     = lane & 15;   // row within tile / N index
    const int nwaves = blockDim.x >> 5;

    float t0 = NEGBIG, t1 = NEGBIG;   // running top-2 of squared distances

    for (int p = wid; p < npairs; p += nwaves) {
        // decode p -> (ti, tj) with ti <= tj
        int pp = p, ti = 0;
        while (pp >= (ntiles - ti)) { pp -= (ntiles - ti); ++ti; }
        int tj = ti + pp;

        int rowA = members[cbase + ti * 16 + lm];
        int rowB = members[cbase + tj * 16 + lm];
        const float* pa = y + (size_t)rowA * D;
        const float* pb = y + (size_t)rowB * D;

        v8f acc = {};
        for (int k0 = 0; k0 < D; k0 += 4) {
            int ks = k0 + 2 * half;
            float2 av = *(const float2*)(pa + ks);
            float2 bv = *(const float2*)(pb + ks);
            v2f a = { av.x, av.y };
            v2f b = { bv.x, bv.y };
            acc = __builtin_amdgcn_wmma_f32_16x16x4_f32(
                false, a, false, b, (short)0, acc, false, false);
        }

        // C/D layout: VGPR r, lanes 0-15 -> (M=r, N=lane); 16-31 -> (M=r+8)
        for (int r = 0; r < 8; ++r) {
            int M     = r + 8 * half;
            int aSlot = ti * 16 + M;
            int bSlot = tj * 16 + lm;
            float d2 = NEGBIG;
            if (aSlot < m && bSlot < m && aSlot < bSlot) {
                int gA = members[cbase + aSlot];
                int gB = members[cbase + bSlot];
                d2 = fmaxf(sq[gA] + sq[gB] - 2.0f * acc[r], 0.0f);
            }
            if (d2 > t0)      { t1 = t0; t0 = d2; }
            else if (d2 > t1) { t1 = d2; }
        }
    }

    // wave32 top-2 merge: top2({a0,a1} U {b0,b1}) =
    //   n0 = max(a0,b0); n1 = max(min(a0,b0), max(a1,b1))
    for (int off = 16; off > 0; off >>= 1) {
        float o0 = __shfl_xor(t0, off, 32);
        float o1 = __shfl_xor(t1, off, 32);
        float n0 = fmaxf(t0, o0);
        float n1 = fmaxf(fminf(t0, o0), fmaxf(t1, o1));
        t0 = n0; t1 = n1;
    }

    __shared__ float stop[2 * 8];
    if (lane == 0) { stop[wid * 2] = t0; stop[wid * 2 + 1] = t1; }
    __syncthreads();

    if (tid == 0) {
        float a0 = NEGBIG, a1 = NEGBIG;
        for (int w = 0; w < nwaves; ++w) {
            float b0 = stop[w * 2], b1 = stop[w * 2 + 1];
            float n0 = fmaxf(a0, b0);
            float n1 = fmaxf(fminf(a0, b0), fmaxf(a1, b1));
            a0 = n0; a1 = n1;
        }
        // reference: top-2 of sqrt distances; K/(-inf sum) contributes -0
        float term = 0.0f;
        if (a1 > 0.5f * NEGBIG) {
            term = (float)KTOP / (sqrtf(a0) + sqrtf(a1));
        }
        terms[c] = term;
    }
}

// --- kernel 5: center min-distance + final combine -------------------------
__global__ __launch_bounds__(256) void final_kernel(const float* __restrict__ centers,
                                                    const float* __restrict__ terms,
                                                    float* __restrict__ out,
                                                    int D) {
    __shared__ float smin[256];
    int tid = threadIdx.x;
    const int npairs = NCLS * (NCLS - 1) / 2;   // 496

    float localmin = 3.0e38f;
    for (int p = tid; p < npairs; p += blockDim.x) {
        int pp = p, i = 0;
        while (pp >= (NCLS - 1 - i)) { pp -= (NCLS - 1 - i); ++i; }
        int j = i + 1 + pp;
        float acc = 0.0f;
        const float* ci = centers + (size_t)i * D;
        const float* cj = centers + (size_t)j * D;
        for (int d = 0; d < D; ++d) {
            float diff = ci[d] - cj[d];
            acc = fmaf(diff, diff, acc);
        }
        localmin = fminf(localmin, acc);
    }
    smin[tid] = localmin;
    __syncthreads();
    for (int s = (int)blockDim.x / 2; s > 0; s >>= 1) {
        if (tid < s) smin[tid] = fminf(smin[tid], smin[tid + s]);
        __syncthreads();
    }
    if (tid == 0) {
        float lsum = 0.0f;
        for (int c = 0; c < NCLS; ++c) lsum += terms[c];
        float dmin = sqrtf(fmaxf(smin[0], 0.0f));
        const float MARGIN = 10.0f, ALPHA = 1.0f, BETA = 0.5f;
        out[0] = ALPHA * lsum + BETA * fmaxf(MARGIN - dmin, 0.0f);
    }
}

// ---------------------------------------------------------------------------
extern "C" void kernel_launch(void* const* d_in, const int* in_sizes, int n_in,
                              void* d_out, int out_size, void* d_ws, size_t ws_size,
                              hipStream_t stream) {
    const float* y   = (const float*)d_in[0];
    const int*   tgt = (const int*)d_in[1];
    float*       out = (float*)d_out;

    const int B = in_sizes[1];            // 2048
    const int D = in_sizes[0] / B;        // 512

    // workspace layout (all 4B-aligned): counts | members | sq | centers | terms
    char*  ws      = (char*)d_ws;
    int*   counts  = (int*)ws;                                   // NCLS ints
    int*   members = (int*)(ws + 256);                           // NCLS*MAXM ints
    float* sq      = (float*)(ws + 256 + (size_t)NCLS * MAXM * sizeof(int)); // B floats
    float* centers = sq + B;                                     // NCLS*D floats
    float* terms   = centers + (size_t)NCLS * D;                 // NCLS floats

    bucket_kernel<<<1, 32, 0, stream>>>(tgt, counts, members, B);
    rownorm_kernel<<<(B * 32 + 255) / 256, 256, 0, stream>>>(y, sq, B, D);
    centers_kernel<<<NCLS, 256, 0, stream>>>(y, members, counts, centers, D);
    intra_topk_kernel<<<NCLS, 256, 0, stream>>>(y, members, counts, sq, terms, D);
    final_kernel<<<1, 256, 0, stream>>>(centers, terms, out, D);
}